// ImprovedGCNWithPositionalEncoding_81973745812106
// MI455X (gfx1250) — compile-verified
//
#include <hip/hip_runtime.h>
#include <hip/hip_bf16.h>

// ---------------------------------------------------------------------------
// Types for WMMA f32 16x16x4 (CDNA5 / gfx1250, wave32)
// A: 16x4 f32  -> 2 VGPRs/lane   (lanes 0-15: K0,K1 ; lanes 16-31: K2,K3; M = lane%16)
// B: 4x16 f32  -> 2 VGPRs/lane   (lanes 0-15: K0,K1 ; lanes 16-31: K2,K3; N = lane%16)
// C/D: 16x16   -> 8 VGPRs/lane   (vgpr v: M = v + 8*(lane/16), N = lane%16)
// ---------------------------------------------------------------------------
typedef __attribute__((ext_vector_type(2))) float v2f;
typedef __attribute__((ext_vector_type(8))) float v8f;

#define HID 64
#define IN_DIM 128
#define POS_DIM 64
#define KIN 192   // IN_DIM + POS_DIM

// ---------------------------------------------------------------------------
// Sinusoidal positional encoding:  pe[n][2j] = sin(pos * e^{-2j ln(1e4)/64}),
//                                  pe[n][2j+1] = cos(...)
// ---------------------------------------------------------------------------
__global__ void pe_kernel(const int* __restrict__ positions, float* __restrict__ pe, int n) {
    int g = blockIdx.x * blockDim.x + threadIdx.x;
    if (g >= n * POS_DIM) return;
    int nid = g >> 6;
    int d   = g & 63;
    float pos = (float)positions[nid];
    int j = d >> 1;
    // exp(2j * (-ln(10000)/64)) = exp(-j * 0.287823136624...)
    float div = expf(-0.28782313662425572f * (float)j);
    float ang = pos * div;
    pe[g] = (d & 1) ? cosf(ang) : sinf(ang);
}

__global__ void zero_kernel(float* __restrict__ p, int count) {
    int g = blockIdx.x * blockDim.x + threadIdx.x;
    if (g < count) p[g] = 0.0f;
}

// deg[c] += ew[e]  (then later dinv = rsqrt(deg + 1))
__global__ void deg_kernel(const int* __restrict__ col, const float* __restrict__ ew,
                           float* __restrict__ deg, int E) {
    int e = blockIdx.x * blockDim.x + threadIdx.x;
    if (e < E) atomicAdd(&deg[col[e]], ew[e]);
}

__global__ void dinv_kernel(float* __restrict__ dinv, int n) {
    int g = blockIdx.x * blockDim.x + threadIdx.x;
    if (g < n) dinv[g] = rsqrtf(dinv[g] + 1.0f);
}

// ---------------------------------------------------------------------------
// Input GEMM:  t[N,64] = concat(x[N,128], pe[N,64]) @ ft_W[192,64] + ft_b
// Block = 256 threads = 8 waves; wave w computes rows [blk*128 + 16w, +16).
// ft_W staged in LDS (48 KB).
// ---------------------------------------------------------------------------
__global__ void input_gemm_kernel(const float* __restrict__ x, const float* __restrict__ pe,
                                  const float* __restrict__ ftW, const float* __restrict__ ftb,
                                  float* __restrict__ out, int n) {
    __shared__ float Wl[KIN * HID];   // 192*64*4 = 48 KB
    int tid = threadIdx.x;
    for (int i = tid; i < KIN * HID; i += 256) Wl[i] = ftW[i];
    __syncthreads();

    int wave = tid >> 5, lane = tid & 31;
    int rowBase = blockIdx.x * 128 + wave * 16;
    int m = lane & 15;                // M (and N) index within tile
    int khalf = (lane >> 4) * 2;      // 0 or 2

    v8f acc[4] = {};                  // four 16x16 N-tiles covering HID=64

    int row = rowBase + m;
    bool inb = row < n;
    const float* xr  = x  + (long)(inb ? row : 0) * IN_DIM;
    const float* per = pe + (long)(inb ? row : 0) * POS_DIM;

    for (int kk = 0; kk < KIN / 4; ++kk) {
        int kb = kk * 4;
        int k0 = kb + khalf;
        v2f a;
        float a0 = (k0     < IN_DIM) ? xr[k0]     : per[k0 - IN_DIM];
        float a1 = (k0 + 1 < IN_DIM) ? xr[k0 + 1] : per[k0 + 1 - IN_DIM];
        a.x = inb ? a0 : 0.0f;
        a.y = inb ? a1 : 0.0f;
#pragma unroll
        for (int t = 0; t < 4; ++t) {
            v2f b;
            b.x = Wl[(k0)     * HID + t * 16 + m];
            b.y = Wl[(k0 + 1) * HID + t * 16 + m];
            acc[t] = __builtin_amdgcn_wmma_f32_16x16x4_f32(
                false, a, false, b, (short)0, acc[t], false, false);
        }
    }

    int rb = rowBase + (lane >> 4) * 8;
#pragma unroll
    for (int t = 0; t < 4; ++t) {
        float bias = ftb[t * 16 + m];
#pragma unroll
        for (int v = 0; v < 8; ++v) {
            int r = rb + v;
            if (r < n) out[(long)r * HID + t * 16 + m] = acc[t][v] + bias;
        }
    }
}

// ---------------------------------------------------------------------------
// Layer GEMM:  hw[N,64] = h[N,64] @ W[64,64]   (bias applied later)
// ---------------------------------------------------------------------------
__global__ void gemm64_kernel(const float* __restrict__ A, const float* __restrict__ W,
                              float* __restrict__ out, int n) {
    __shared__ float Wl[HID * HID];   // 16 KB
    int tid = threadIdx.x;
    for (int i = tid; i < HID * HID; i += 256) Wl[i] = W[i];
    __syncthreads();

    int wave = tid >> 5, lane = tid & 31;
    int rowBase = blockIdx.x * 128 + wave * 16;
    int m = lane & 15;
    int khalf = (lane >> 4) * 2;

    v8f acc[4] = {};

    int row = rowBase + m;
    bool inb = row < n;
    const float* arow = A + (long)(inb ? row : 0) * HID;

    for (int kk = 0; kk < HID / 4; ++kk) {
        int k0 = kk * 4 + khalf;
        v2f a;
        a.x = inb ? arow[k0]     : 0.0f;
        a.y = inb ? arow[k0 + 1] : 0.0f;
#pragma unroll
        for (int t = 0; t < 4; ++t) {
            v2f b;
            b.x = Wl[(k0)     * HID + t * 16 + m];
            b.y = Wl[(k0 + 1) * HID + t * 16 + m];
            acc[t] = __builtin_amdgcn_wmma_f32_16x16x4_f32(
                false, a, false, b, (short)0, acc[t], false, false);
        }
    }

    int rb = rowBase + (lane >> 4) * 8;
#pragma unroll
    for (int t = 0; t < 4; ++t)
#pragma unroll
        for (int v = 0; v < 8; ++v) {
            int r = rb + v;
            if (r < n) out[(long)r * HID + t * 16 + m] = acc[t][v];
        }
}

// ---------------------------------------------------------------------------
// Edge scatter: agg[col[e]][c] += dinv[row]*ew*dinv[col] * hw[row[e]][c]
// One thread per (edge, channel): lanes hit consecutive addresses.
// ---------------------------------------------------------------------------
__global__ void scatter_kernel(const float* __restrict__ hw, const int* __restrict__ row,
                               const int* __restrict__ col, const float* __restrict__ ew,
                               const float* __restrict__ dinv, float* __restrict__ agg, int E) {
    int g = blockIdx.x * blockDim.x + threadIdx.x;
    if (g >= E * HID) return;
    int e = g >> 6;
    int c = g & 63;
    int r  = row[e];
    int cl = col[e];
    float norm = dinv[r] * ew[e] * dinv[cl];
    atomicAdd(&agg[(long)cl * HID + c], norm * hw[(long)r * HID + c]);
}

// t = agg + dinv^2 * hw + bias
__global__ void selfloop_kernel(const float* __restrict__ agg, const float* __restrict__ hw,
                                const float* __restrict__ dinv, const float* __restrict__ bias,
                                float* __restrict__ t, int n) {
    int g = blockIdx.x * blockDim.x + threadIdx.x;
    if (g >= n * HID) return;
    int nid = g >> 6;
    int c   = g & 63;
    float di = dinv[nid];
    t[g] = agg[g] + di * di * hw[g] + bias[c];
}

// ---------------------------------------------------------------------------
// Column statistics for BN: stats[0:64]=sum, stats[64:128]=sum of squares
// ---------------------------------------------------------------------------
__global__ void colstats_kernel(const float* __restrict__ t, float* __restrict__ stats, int total) {
    __shared__ float ssum[256];
    __shared__ float ssq[256];
    int tid = threadIdx.x;
    float s = 0.0f, q = 0.0f;
    for (int g = blockIdx.x * 256 + tid; g < total; g += gridDim.x * 256) {
        float v = t[g];
        s += v;
        q += v * v;
    }
    ssum[tid] = s;
    ssq[tid]  = q;
    __syncthreads();
    if (tid < 64) {
        s = ssum[tid] + ssum[tid + 64] + ssum[tid + 128] + ssum[tid + 192];
        q = ssq[tid]  + ssq[tid + 64]  + ssq[tid + 128]  + ssq[tid + 192];
        atomicAdd(&stats[tid], s);
        atomicAdd(&stats[64 + tid], q);
    }
}

// stats[128:192]=scale, stats[192:256]=shift  (biased variance, as reference)
__global__ void bnfin_kernel(float* __restrict__ stats, const float* __restrict__ gamma,
                             const float* __restrict__ beta, float invn) {
    int c = threadIdx.x;
    if (c >= 64) return;
    float mean = stats[c] * invn;
    float var  = stats[64 + c] * invn - mean * mean;
    float sc   = gamma[c] * rsqrtf(var + 1e-5f);
    stats[128 + c] = sc;
    stats[192 + c] = beta[c] - mean * sc;
}

// h = relu(t*scale + shift) [+ h if addres]
__global__ void bnrelu_kernel(const float* __restrict__ t, const float* __restrict__ stats,
                              float* __restrict__ h, int n, int addres) {
    int g = blockIdx.x * blockDim.x + threadIdx.x;
    if (g >= n * HID) return;
    int c = g & 63;
    float v = fmaxf(t[g] * stats[128 + c] + stats[192 + c], 0.0f);
    h[g] = addres ? (h[g] + v) : v;
}

// ---------------------------------------------------------------------------
// Output head: out[n] = relu(h[n] @ W1 + b1) @ W2 + b2
// ---------------------------------------------------------------------------
__global__ void out_kernel(const float* __restrict__ h, const float* __restrict__ W1,
                           const float* __restrict__ b1, const float* __restrict__ W2,
                           const float* __restrict__ b2, float* __restrict__ out, int n) {
    __shared__ float w1[HID * 32];   // 8 KB
    __shared__ float b1s[32];
    __shared__ float w2[32];
    int tid = threadIdx.x;
    for (int i = tid; i < HID * 32; i += 256) w1[i] = W1[i];
    if (tid < 32) { b1s[tid] = b1[tid]; w2[tid] = W2[tid]; }
    __syncthreads();

    int nid = blockIdx.x * 256 + tid;
    if (nid >= n) return;
    const float* hr = h + (long)nid * HID;
    float hk[HID];
#pragma unroll
    for (int k = 0; k < HID; ++k) hk[k] = hr[k];

    float acc2 = b2[0];
    for (int j = 0; j < 32; ++j) {
        float s = b1s[j];
#pragma unroll
        for (int k = 0; k < HID; ++k) s += hk[k] * w1[k * 32 + j];
        acc2 += fmaxf(s, 0.0f) * w2[j];
    }
    out[nid] = acc2;
}

// ---------------------------------------------------------------------------
// Launcher
// ---------------------------------------------------------------------------
extern "C" void kernel_launch(void* const* d_in, const int* in_sizes, int n_in,
                              void* d_out, int out_size, void* d_ws, size_t ws_size,
                              hipStream_t stream) {
    const float* x         = (const float*)d_in[0];
    const int*   ei        = (const int*)  d_in[1];
    const float* ew        = (const float*)d_in[2];
    const int*   positions = (const int*)  d_in[3];
    const float* ft_W      = (const float*)d_in[4];
    const float* ft_b      = (const float*)d_in[5];
    const float* ft_gamma  = (const float*)d_in[6];
    const float* ft_beta   = (const float*)d_in[7];
    const float* conv_W    = (const float*)d_in[8];
    const float* conv_b    = (const float*)d_in[9];
    const float* bn_gamma  = (const float*)d_in[10];
    const float* bn_beta   = (const float*)d_in[11];
    const float* out_W1    = (const float*)d_in[12];
    const float* out_b1    = (const float*)d_in[13];
    const float* out_W2    = (const float*)d_in[14];
    const float* out_b2    = (const float*)d_in[15];

    const int N = in_sizes[0] / IN_DIM;
    const int E = in_sizes[2];

    const int* row = ei;
    const int* col = ei + E;

    // Workspace layout (floats)
    float* ws    = (float*)d_ws;
    float* pe    = ws;                 // N*64
    float* h     = pe   + (long)N * 64;
    float* t     = h    + (long)N * 64;
    float* hw    = t    + (long)N * 64;
    float* agg   = hw   + (long)N * 64;
    float* dinv  = agg  + (long)N * 64;   // N
    float* stats = dinv + N;              // 256

    const int NC   = N * HID;                         // N*64 elements
    const int gNC  = (NC + 255) / 256;
    const int gN   = (N + 255) / 256;
    const int gE   = (E + 255) / 256;
    const int gEC  = (E * HID + 255) / 256;           // E*64 threads
    const int gGEMM = (N + 127) / 128;                // 128 rows / block
    const float invn = 1.0f / (float)N;

    // 1. positional encoding table
    pe_kernel<<<gNC, 256, 0, stream>>>(positions, pe, N);

    // 2. degrees -> dinv (layer invariant)
    zero_kernel<<<gN, 256, 0, stream>>>(dinv, N);
    deg_kernel<<<gE, 256, 0, stream>>>(col, ew, dinv, E);
    dinv_kernel<<<gN, 256, 0, stream>>>(dinv, N);

    // 3. feature transform: t = concat(x, pe) @ ft_W + ft_b
    input_gemm_kernel<<<gGEMM, 256, 0, stream>>>(x, pe, ft_W, ft_b, t, N);

    // 4. BN + ReLU -> h
    zero_kernel<<<1, 128, 0, stream>>>(stats, 128);
    colstats_kernel<<<512, 256, 0, stream>>>(t, stats, NC);
    bnfin_kernel<<<1, 64, 0, stream>>>(stats, ft_gamma, ft_beta, invn);
    bnrelu_kernel<<<gNC, 256, 0, stream>>>(t, stats, h, N, 0);

    // 5. GCN layers
    for (int i = 0; i < 3; ++i) {
        gemm64_kernel<<<gGEMM, 256, 0, stream>>>(h, conv_W + (long)i * HID * HID, hw, N);
        zero_kernel<<<gNC, 256, 0, stream>>>(agg, NC);
        scatter_kernel<<<gEC, 256, 0, stream>>>(hw, row, col, ew, dinv, agg, E);
        selfloop_kernel<<<gNC, 256, 0, stream>>>(agg, hw, dinv, conv_b + (long)i * HID, t, N);
        zero_kernel<<<1, 128, 0, stream>>>(stats, 128);
        colstats_kernel<<<512, 256, 0, stream>>>(t, stats, NC);
        bnfin_kernel<<<1, 64, 0, stream>>>(stats, bn_gamma + (long)i * HID, bn_beta + (long)i * HID, invn);
        bnrelu_kernel<<<gNC, 256, 0, stream>>>(t, stats, h, N, 1);
    }

    // 6. MLP head
    out_kernel<<<gN, 256, 0, stream>>>(h, out_W1, out_b1, out_W2, out_b2, (float*)d_out, N);
}